// SkinDeformNet_2800318677121
// MI455X (gfx1250) — compile-verified
//
#include <hip/hip_runtime.h>
#include <hip/hip_bf16.h>

// ---------------------------------------------------------------------------
// SMPL-style LBS on gfx1250.
//   out3 [N,3] | T [N,16] | Rs [B,24,9] | newJ [B,24,3]   (flat in d_out)
// Skinning T = ws @ A16[batch] done with chained V_WMMA_F32_16X16X4_F32
// (K=24 -> 6 wmma per 16-point tile), fp32 end-to-end like the reference.
// ---------------------------------------------------------------------------

typedef __attribute__((ext_vector_type(2))) float v2f;
typedef __attribute__((ext_vector_type(8))) float v8f;

#define N_PTS   440960
#define NBATCH  64
#define NTILES  (N_PTS / 16)      /* 27560 exact */
#define WAVES   8                 /* waves per block in skin kernel */

__constant__ int d_parents[24] = {-1, 0, 0, 0, 1, 2, 3, 4, 5, 6, 7, 8,
                                   9, 9, 9, 12, 13, 14, 16, 17, 18, 19, 20, 21};

// -------------------------------------------------------------------- stage 1
__global__ void rodrigues_kernel(const float* __restrict__ poses,
                                 float* __restrict__ Rs) {
  int t = blockIdx.x * 256 + threadIdx.x;          // t < 1536 (grid exact)
  const float* r = poses + t * 3;
  float x = r[0], y = r[1], z = r[2];
  float ang = sqrtf(x * x + y * y + z * z) + 1e-8f;
  float inv = 1.0f / ang;
  float ax = x * inv, ay = y * inv, az = z * inv;
  float s, co;
  sincosf(ang, &s, &co);
  float oc = 1.0f - co;
  float* R = Rs + t * 9;
  R[0] = 1.0f + oc * (-(ay * ay + az * az));
  R[1] = -s * az + oc * (ax * ay);
  R[2] =  s * ay + oc * (ax * az);
  R[3] =  s * az + oc * (ax * ay);
  R[4] = 1.0f + oc * (-(ax * ax + az * az));
  R[5] = -s * ax + oc * (ay * az);
  R[6] = -s * ay + oc * (ax * az);
  R[7] =  s * ax + oc * (ay * az);
  R[8] = 1.0f + oc * (-(ax * ax + ay * ay));
}

// -------------------------------------------------------------------- stage 2
// One single-wave (32-thread) block per batch element; chain lives in LDS.
// Barriers in a single-wave workgroup lower to S_NOP.
__global__ void chain_kernel(const float* __restrict__ Js,   // [B,24,3]
                             const float* __restrict__ Rs,   // [B,24,9]
                             float* __restrict__ A16,        // [B,24,16] ws
                             float* __restrict__ newJ) {     // [B,24,3]
  __shared__ float chain[24][16];
  __shared__ float M[16];
  int b = blockIdx.x;
  int l = threadIdx.x;
  const float* Jb = Js + b * 72;
  const float* Rb = Rs + b * 216;

  if (l < 16) {                    // joint 0: make(R0, J0)
    int r = l >> 2, c = l & 3;
    float v;
    if (r == 3)      v = (c == 3) ? 1.0f : 0.0f;
    else if (c == 3) v = Jb[r];
    else             v = Rb[r * 3 + c];
    chain[0][l] = v;
  }
  __syncthreads();

  for (int i = 1; i < 24; ++i) {
    int p = d_parents[i];
    if (l < 16) {
      int r = l >> 2, c = l & 3;
      float v;
      if (r == 3)      v = (c == 3) ? 1.0f : 0.0f;
      else if (c == 3) v = Jb[i * 3 + r] - Jb[p * 3 + r];
      else             v = Rb[i * 9 + r * 3 + c];
      M[l] = v;
    }
    __syncthreads();
    if (l < 16) {
      int r = l >> 2, c = l & 3;
      float acc = 0.0f;
#pragma unroll
      for (int q = 0; q < 4; ++q) acc += chain[p][r * 4 + q] * M[q * 4 + c];
      chain[i][l] = acc;
    }
    __syncthreads();
  }

  for (int idx = l; idx < 72; idx += 32) {      // new_J = chain[:, :3, 3]
    int j = idx / 3, r = idx - j * 3;
    newJ[b * 72 + idx] = chain[j][r * 4 + 3];
  }
  for (int idx = l; idx < 384; idx += 32) {     // A = chain, col3 -= R_world@J
    int j = idx >> 4, e = idx & 15, r = e >> 2, c = e & 3;
    float v = chain[j][e];
    if (c == 3) {
      v -= chain[j][r * 4 + 0] * Jb[j * 3 + 0] +
           chain[j][r * 4 + 1] * Jb[j * 3 + 1] +
           chain[j][r * 4 + 2] * Jb[j * 3 + 2];
    }
    A16[b * 384 + idx] = v;
  }
}

// -------------------------------------------------------------------- stage 3
// One wave per 16-point tile. batch is sorted, so tiles are batch-uniform in
// all but <64 of 27560 cases -> WMMA fast path, scalar fallback otherwise.
__global__ void __launch_bounds__(256) skin_kernel(
    const float* __restrict__ ps,    // [N,3]
    const float* __restrict__ ws,    // [N,24]
    const int*   __restrict__ batch, // [N]
    const float* __restrict__ A16,   // [B,24,16]
    float* __restrict__ out3,        // [N,3]
    float* __restrict__ Tout) {      // [N,16]
  __shared__ float Tlds[WAVES][16][17];

  int wave  = threadIdx.x >> 5;
  int lane  = threadIdx.x & 31;
  int tile  = blockIdx.x * WAVES + wave;          // grid exact: 3445*8 tiles
  int n0    = tile << 4;
  int row   = lane & 15;
  int khalf = lane >> 4;

  // Force scalar (wave-uniform) branch so EXEC stays all-ones for WMMA.
  int b0 = __builtin_amdgcn_readfirstlane(batch[n0]);
  int b1 = __builtin_amdgcn_readfirstlane(batch[n0 + 15]);

  if (b0 == b1) {
    // ---- WMMA path: D(16x16) = sum_kc A(16x4) * B(4x16), K = 24 ----
    const float* wsrow = ws + (size_t)(n0 + row) * 24;
    const float* Ab    = A16 + b0 * 384;
    v8f c = {0.f, 0.f, 0.f, 0.f, 0.f, 0.f, 0.f, 0.f};
#pragma unroll
    for (int kc = 0; kc < 6; ++kc) {
      int k0 = kc * 4 + khalf * 2;                // A/B VGPR lane layout: K0/K2
      v2f a, bm;
      float2 wv = *(const float2*)(wsrow + k0);   // 8B-aligned (k0 even)
      a[0]  = wv.x;
      a[1]  = wv.y;
      bm[0] = Ab[k0 * 16 + row];                  // B row k0,   col = row
      bm[1] = Ab[(k0 + 1) * 16 + row];            // B row k0+1, col = row
      c = __builtin_amdgcn_wmma_f32_16x16x4_f32(
          /*neg_a=*/false, a, /*neg_b=*/false, bm,
          /*c_mod=*/(short)0, c, /*reuse_a=*/false, /*reuse_b=*/false);
    }
    // D layout: VGPR v -> rows v / v+8, col = lane%16. Coalesced T stores.
#pragma unroll
    for (int v = 0; v < 8; ++v) {
      int m = v + 8 * khalf;
      Tout[(size_t)(n0 + m) * 16 + row] = c[v];
      Tlds[wave][m][row] = c[v];
    }
  } else {
    // ---- rare boundary tile: per-point scalar FMA path ----
    int n  = n0 + row;
    int bn = batch[n];
    const float* wsrow = ws + (size_t)n * 24;
    const float* Ab    = A16 + bn * 384 + khalf * 8;
    float acc[8];
#pragma unroll
    for (int j = 0; j < 8; ++j) acc[j] = 0.0f;
    for (int k = 0; k < 24; ++k) {
      float w = wsrow[k];
      const float* Ak = Ab + k * 16;
#pragma unroll
      for (int j = 0; j < 8; ++j) acc[j] += w * Ak[j];
    }
#pragma unroll
    for (int j = 0; j < 8; ++j) {
      Tout[(size_t)n * 16 + khalf * 8 + j] = acc[j];
      Tlds[wave][row][khalf * 8 + j] = acc[j];
    }
  }

  __syncthreads();   // converged point; makes the LDS tile visible

  if (lane < 16) {   // epilogue: out = T @ [p,1]
    int n = n0 + lane;
    float px = ps[n * 3 + 0], py = ps[n * 3 + 1], pz = ps[n * 3 + 2];
    const float* Tr = Tlds[wave][lane];
    out3[n * 3 + 0] = Tr[0] * px + Tr[1] * py + Tr[2]  * pz + Tr[3];
    out3[n * 3 + 1] = Tr[4] * px + Tr[5] * py + Tr[6]  * pz + Tr[7];
    out3[n * 3 + 2] = Tr[8] * px + Tr[9] * py + Tr[10] * pz + Tr[11];
  }
}

// ---------------------------------------------------------------------------
extern "C" void kernel_launch(void* const* d_in, const int* in_sizes, int n_in,
                              void* d_out, int out_size, void* d_ws, size_t ws_size,
                              hipStream_t stream) {
  const float* ps    = (const float*)d_in[0];
  const float* Js    = (const float*)d_in[1];
  const float* ws    = (const float*)d_in[2];
  const float* poses = (const float*)d_in[3];
  const int*   batch = (const int*)d_in[4];

  float* out = (float*)d_out;
  const size_t OUT3_OFF = 0;
  const size_t T_OFF    = (size_t)N_PTS * 3;                  // 1,322,880
  const size_t RS_OFF   = T_OFF + (size_t)N_PTS * 16;         // 8,378,240
  const size_t NJ_OFF   = RS_OFF + (size_t)NBATCH * 24 * 9;   // 8,392,064

  float* A16 = (float*)d_ws;                                  // [64][24][16]

  rodrigues_kernel<<<6, 256, 0, stream>>>(poses, out + RS_OFF);
  chain_kernel<<<NBATCH, 32, 0, stream>>>(Js, out + RS_OFF, A16, out + NJ_OFF);
  skin_kernel<<<NTILES / WAVES, 256, 0, stream>>>(ps, ws, batch, A16,
                                                  out + OUT3_OFF, out + T_OFF);
}